// StrideHDEncoder_68762426409218
// MI455X (gfx1250) — compile-verified
//
#include <hip/hip_runtime.h>
#include <hip/hip_bf16.h>

// ---------------------------------------------------------------------------
// StrideHDEncoder for MI455X (gfx1250, wave32)
//
// Pipeline:
//   K0: init per-batch min/max slots (flipped-uint encoding) in d_ws
//   K1: max-pool 8x8 s4 per (b,c,oh) row, LDS-staged via async global->LDS
//       (ASYNCcnt path), wave32 shuffle reduce -> 2 atomics/block for batch
//       min/max
//   K2: per batch row: zero 8192 outputs, recompute the <=512 gathered pooled
//       windows from x (L2-resident), pack bits -> p, scatter 64 ones.
// ---------------------------------------------------------------------------

#define NB       256
#define NC       3
#define NH       224
#define NW       224
#define OHW      55          // (224-8)/4 + 1
#define PLANE    (NH * NW)   // 50176
#define FEATS    (NC * OHW * OHW)          // 9075
#define NELEM    (FEATS * 8)               // 72600
#define MROWS    64
#define VECSZ    128
#define OUTD     8192
#define ROWF     (8 * NW)    // 1792 floats staged per pooling block

#ifndef __has_builtin
#define __has_builtin(x) 0
#endif

#if __has_builtin(__builtin_amdgcn_global_load_async_to_lds_b128) && \
    __has_builtin(__builtin_amdgcn_s_wait_asynccnt)
#define USE_ASYNC_LDS 1
// Builtin signature (from clang diagnostic): param0 is
//   'int __attribute__((vector_size(16))) __device__ *'  == addrspace(1) int4*
// so param1 is the LDS-side int4* in addrspace(3).
typedef int v4i_t __attribute__((ext_vector_type(4)));
typedef __attribute__((address_space(1))) v4i_t* as1_v4i_p;
typedef __attribute__((address_space(3))) v4i_t* as3_v4i_p;
#else
#define USE_ASYNC_LDS 0
#endif

// Order-preserving float <-> uint map so we can use integer atomicMin/Max.
__device__ __forceinline__ unsigned flipf(float f) {
    unsigned u = __float_as_uint(f);
    return (u & 0x80000000u) ? ~u : (u | 0x80000000u);
}
__device__ __forceinline__ float unflipf(unsigned u) {
    unsigned v = (u & 0x80000000u) ? (u ^ 0x80000000u) : ~u;
    return __uint_as_float(v);
}

// ---------------------------------------------------------------------------
__global__ __launch_bounds__(256) void hd_init_minmax(unsigned* __restrict__ minU,
                                                      unsigned* __restrict__ maxU) {
    int t = threadIdx.x;            // 256 threads, 256 batches
    minU[t] = 0xFFFFFFFFu;          // flip(+inf-ish): max uint
    maxU[t] = 0u;                   // min uint
}

// ---------------------------------------------------------------------------
// One block per (oh, c, b): 64 threads (2 wave32), 7168 B LDS tile.
__global__ __launch_bounds__(64) void hd_pool_minmax(const float* __restrict__ x,
                                                     unsigned* __restrict__ minU,
                                                     unsigned* __restrict__ maxU) {
    __shared__ float tile[ROWF];    // 8 contiguous input rows of one plane
    __shared__ float wmin[2], wmax[2];

    const int oh = blockIdx.x;      // 0..54
    const int c  = blockIdx.y;      // 0..2
    const int b  = blockIdx.z;      // 0..255
    const int tid = threadIdx.x;

    // Rows ih = 4*oh .. 4*oh+7 are contiguous in memory: one 1792-float block.
    const float* src = x + ((size_t)(b * NC + c) * NH + (size_t)oh * 4) * NW;

    // Hint the next vertical band into cache (emits global_prefetch_b8).
    __builtin_prefetch(src + ROWF, 0, 1);

#if USE_ASYNC_LDS
    {
        const char* g = (const char*)src;
        const unsigned lbase = (unsigned)(unsigned long long)(uintptr_t)(void*)tile;
        #pragma unroll
        for (int i = 0; i < 7; ++i) {               // 7 * 64 lanes * 16B = 7168B
            const unsigned off = (unsigned)(i * 64 + tid) * 16u;
            as1_v4i_p gp = (as1_v4i_p)(unsigned long long)(uintptr_t)(g + off);
            as3_v4i_p lp = (as3_v4i_p)(lbase + off);
            __builtin_amdgcn_global_load_async_to_lds_b128(gp, lp, 0, 0);
        }
        __builtin_amdgcn_s_wait_asynccnt(0);
    }
#else
    {
        const float4* s4 = (const float4*)src;      // 896B row pitch -> 16B aligned
        float4* t4 = (float4*)tile;
        #pragma unroll
        for (int i = 0; i < 7; ++i)                 // 448 float4 total
            t4[tid + i * 64] = s4[tid + i * 64];
    }
#endif
    __syncthreads();

    float vmin =  __builtin_inff();
    float vmax = -__builtin_inff();
    if (tid < OHW) {
        const float* w = tile + tid * 4;            // window cols 4*ow .. 4*ow+7
        float m = -__builtin_inff();
        #pragma unroll
        for (int r = 0; r < 8; ++r) {
            const float* row = w + r * NW;
            #pragma unroll
            for (int k = 0; k < 8; ++k) m = fmaxf(m, row[k]);
        }
        vmin = m;
        vmax = m;
    }

    // wave32 tree reduce
    #pragma unroll
    for (int off = 16; off > 0; off >>= 1) {
        vmin = fminf(vmin, __shfl_xor(vmin, off, 32));
        vmax = fmaxf(vmax, __shfl_xor(vmax, off, 32));
    }
    if ((tid & 31) == 0) { wmin[tid >> 5] = vmin; wmax[tid >> 5] = vmax; }
    __syncthreads();
    if (tid == 0) {
        float mn = fminf(wmin[0], wmin[1]);
        float mx = fmaxf(wmax[0], wmax[1]);
        atomicMin(&minU[b], flipf(mn));
        atomicMax(&maxU[b], flipf(mx));
    }
}

// ---------------------------------------------------------------------------
// One block per batch: zero the 8192-float row, then 64 threads scatter ones.
__global__ __launch_bounds__(256) void hd_encode(const float* __restrict__ x,
                                                 const int* __restrict__ R,
                                                 const unsigned* __restrict__ minU,
                                                 const unsigned* __restrict__ maxU,
                                                 float* __restrict__ out) {
    const int b = blockIdx.x;
    float* row = out + (size_t)b * OUTD;

    // Zero the row: 2048 float4 across 256 threads.
    float4* row4 = (float4*)row;
    const float4 z = make_float4(0.f, 0.f, 0.f, 0.f);
    #pragma unroll
    for (int i = 0; i < 8; ++i) row4[threadIdx.x + i * 256] = z;
    __syncthreads();

    if (threadIdx.x < MROWS) {
        const float mn  = unflipf(minU[b]);
        const float mx  = unflipf(maxU[b]);
        const float inv = 1.0f / (mx - mn + 1e-8f);
        const int m = threadIdx.x;

        int p = 0;
        #pragma unroll
        for (int k = 0; k < 8; ++k) {
            int idx = R[m * 8 + k] % NELEM;
            if (idx < 0) idx += NELEM;
            const int j  = idx >> 3;        // pooled feature index
            const int t  = idx & 7;         // threshold level
            const int cc  = j / (OHW * OHW);
            const int rem = j - cc * (OHW * OHW);
            const int oh  = rem / OHW;
            const int ow  = rem - oh * OHW;

            const float* src = x + ((size_t)(b * NC + cc) * NH + (size_t)oh * 4) * NW
                                 + (size_t)ow * 4;
            float mval = -__builtin_inff();
            #pragma unroll
            for (int r = 0; r < 8; ++r) {
                const float* rr = src + r * NW;
                #pragma unroll
                for (int q = 0; q < 8; ++q) mval = fmaxf(mval, rr[q]);
            }
            const float xn  = (mval - mn) * inv;
            const float thr = (float)(t + 1) * 0.125f;
            if (xn > thr) p += (128 >> k);  // weights 2^(7-k)
        }
        p &= (VECSZ - 1);                   // p <= 255, so & == % 128
        row[m * VECSZ + p] = 1.0f;
    }
}

// ---------------------------------------------------------------------------
extern "C" void kernel_launch(void* const* d_in, const int* in_sizes, int n_in,
                              void* d_out, int out_size, void* d_ws, size_t ws_size,
                              hipStream_t stream) {
    const float* x = (const float*)d_in[0];   // (256,3,224,224) f32
    const int*   R = (const int*)d_in[1];     // (64,8) i32
    float* out = (float*)d_out;               // (256,8192) f32

    unsigned* minU = (unsigned*)d_ws;         // 256 slots
    unsigned* maxU = minU + NB;               // 256 slots

    hd_init_minmax<<<1, 256, 0, stream>>>(minU, maxU);
    hd_pool_minmax<<<dim3(OHW, NC, NB), 64, 0, stream>>>(x, minU, maxU);
    hd_encode<<<NB, 256, 0, stream>>>(x, R, minU, maxU, out);
}